// Embedder_regression_73151882985825
// MI455X (gfx1250) — compile-verified
//
#include <hip/hip_runtime.h>
#include <hip/hip_bf16.h>

// ---------------------------------------------------------------------------
// GraphSAGE 3-layer inference + global mean pool, CDNA5 (gfx1250).
// Layers: 128->128 (relu), 128->128 (relu), 128->64. Output [64,64] f32.
// ---------------------------------------------------------------------------

#define N_NODES  100000
#define N_EDGES  1600000
#define N_GRAPHS 64
#define F_IN     128   // input feature dim of every layer

typedef __attribute__((ext_vector_type(2))) float v2f;
typedef __attribute__((ext_vector_type(8))) float v8f;

// ---------------------------------------------------------------------------
// zero fill (grid-stride)
// ---------------------------------------------------------------------------
__global__ void zero_kernel(float* __restrict__ p, size_t n) {
    size_t i = (size_t)blockIdx.x * blockDim.x + threadIdx.x;
    size_t stride = (size_t)gridDim.x * blockDim.x;
    for (; i < n; i += stride) p[i] = 0.0f;
}

// ---------------------------------------------------------------------------
// per-node in-degree (float, HW global_atomic_add_f32)
// ---------------------------------------------------------------------------
__global__ void deg_kernel(const int* __restrict__ dst, float* __restrict__ deg, int E) {
    int e = blockIdx.x * blockDim.x + threadIdx.x;
    if (e < E) unsafeAtomicAdd(&deg[dst[e]], 1.0f);
}

// ---------------------------------------------------------------------------
// Pre-pack [Wl;Wr] (256 x DOUT) into K-pair interleaved layout:
//   Wp[((k/2)*DOUT + n)*2 + (k&1)] = Wcomb[k][n]
// so the WMMA B operand (B[kk][n], B[kk+1][n]) is ONE global_load_b64.
// ---------------------------------------------------------------------------
__global__ void pack_weights_kernel(const float* __restrict__ Wl,
                                    const float* __restrict__ Wr,
                                    float* __restrict__ Wp, int dout) {
    int idx = blockIdx.x * blockDim.x + threadIdx.x;
    int total = 2 * F_IN * dout;
    if (idx >= total) return;
    int krow = idx / dout;
    int n    = idx % dout;
    float v = (krow < F_IN) ? Wl[(size_t)krow * dout + n]
                            : Wr[(size_t)(krow - F_IN) * dout + n];
    Wp[((size_t)(krow >> 1) * dout + n) * 2 + (krow & 1)] = v;
}

// ---------------------------------------------------------------------------
// edge scatter: agg[dst] += h[src]. 32 lanes per edge, float4 per lane.
// ---------------------------------------------------------------------------
__global__ void scatter_kernel(const float* __restrict__ h,
                               const int* __restrict__ src,
                               const int* __restrict__ dst,
                               float* __restrict__ agg, int E) {
    int t = blockIdx.x * blockDim.x + threadIdx.x;
    int e = t >> 5;
    if (e >= E) return;
    int lane = t & 31;                      // 32 lanes * 4 floats = 128 feats
    int s = src[e];
    int d = dst[e];
    const float4 v = *(const float4*)(h + (size_t)s * F_IN + lane * 4);
    float* o = agg + (size_t)d * F_IN + lane * 4;
    unsafeAtomicAdd(o + 0, v.x);
    unsafeAtomicAdd(o + 1, v.y);
    unsafeAtomicAdd(o + 2, v.z);
    unsafeAtomicAdd(o + 3, v.w);
}

// ---------------------------------------------------------------------------
// Fused SAGE layer GEMM via V_WMMA_F32_16X16X4_F32:
//   out = [agg*inv_deg | h] (N x 256) @ Wpacked (256 x DOUT) + b (opt. relu)
// Block = 64-row strip (4 M-tiles per wave); wave y owns column tile
// [16y,16y+16). A strip staged in LDS: h-half via global_load_async_to_lds,
// agg-half via VGPRs (needs the 1/deg scale). Row stride 260 dwords:
// 1040 B is 16B-aligned (b128 LDS writes) and ==4 mod 64 banks
// (conflict-free 16-rows-at-one-column-pair WMMA A reads).
// ---------------------------------------------------------------------------
template <int DOUT, bool RELU>
__global__ __launch_bounds__(32 * (DOUT / 16))
void sage_gemm_kernel(const float* __restrict__ agg,
                      const float* __restrict__ hin,
                      const float* __restrict__ deg,
                      const float* __restrict__ Wp,   // packed [128,DOUT,2]
                      const float* __restrict__ bias, // [DOUT]
                      float* __restrict__ out) {
    constexpr int NW   = DOUT / 16;    // waves per block
    constexpr int MT   = 4;            // M tiles per wave
    constexpr int ROWS = 16 * MT;      // 64 rows per block
    constexpr int KT   = 2 * F_IN;     // fused K = 256
    constexpr int RS   = KT + 4;       // padded LDS row stride (dwords)
    constexpr int NT   = NW * 32;
    __shared__ float Asl[ROWS][RS];

    const int m0  = blockIdx.x * ROWS;
    const int tid = threadIdx.y * 32 + threadIdx.x;

    // ---- async-stage h half: LDS cols [128,256) <- hin rows (16B/lane DMA) --
    for (int i = tid; i < ROWS * (F_IN / 4); i += NT) {
        int r  = i >> 5;                       // 32 float4 per row-half
        int c4 = i & 31;
        int m  = min(m0 + r, N_NODES - 1);
        unsigned lds = (unsigned)(uintptr_t)&Asl[r][F_IN + c4 * 4];
        unsigned long long ga =
            (unsigned long long)(uintptr_t)(hin + (size_t)m * F_IN + c4 * 4);
        asm volatile("global_load_async_to_lds_b128 %0, %1, off"
                     :: "v"(lds), "v"(ga) : "memory");
    }
    // ---- stage agg half with 1/max(deg,1) scaling: LDS cols [0,128) --------
    for (int i = tid; i < ROWS * (F_IN / 4); i += NT) {
        int r  = i >> 5;
        int c4 = i & 31;
        int m  = min(m0 + r, N_NODES - 1);
        float4 v = *(const float4*)(agg + (size_t)m * F_IN + c4 * 4);
        float inv = 1.0f / fmaxf(deg[m], 1.0f);
        v.x *= inv; v.y *= inv; v.z *= inv; v.w *= inv;
        *(float4*)(&Asl[r][c4 * 4]) = v;
    }
    asm volatile("s_wait_asynccnt 0x0" ::: "memory");
    __syncthreads();

    // ---- WMMA main loop: 1 b64 weight load feeds 4 independent WMMAs ------
    const int lane = threadIdx.x;
    const int half = lane >> 4;        // f32 A layout: vgpr0 = K+2*half
    const int mr   = lane & 15;
    const int n    = threadIdx.y * 16 + (lane & 15);
    const float* WpBase = Wp + ((size_t)half * DOUT + n) * 2;

    v8f acc[MT] = {};
#pragma unroll 4
    for (int k = 0; k < KT; k += 4) {
        v2f b = *(const v2f*)(WpBase + (size_t)(k >> 1) * DOUT * 2);
#pragma unroll
        for (int t = 0; t < MT; ++t) {
            v2f a = *(const v2f*)(&Asl[16 * t + mr][k + 2 * half]);
            acc[t] = __builtin_amdgcn_wmma_f32_16x16x4_f32(
                /*neg_a=*/false, a, /*neg_b=*/false, b,
                /*c_mod=*/(short)0, acc[t], /*reuse_a=*/false, /*reuse_b=*/false);
        }
    }

    // ---- epilogue: bias (+relu), store -------------------------------------
    float bn = bias[n];
    if (m0 + ROWS <= N_NODES) {
#pragma unroll
        for (int t = 0; t < MT; ++t)
#pragma unroll
            for (int v = 0; v < 8; ++v) {
                float val = acc[t][v] + bn;
                if (RELU) val = fmaxf(val, 0.0f);
                out[(size_t)(m0 + 16 * t + v + 8 * half) * DOUT + n] = val;
            }
    } else {
#pragma unroll
        for (int t = 0; t < MT; ++t)
#pragma unroll
            for (int v = 0; v < 8; ++v) {
                int m = m0 + 16 * t + v + 8 * half;
                if (m < N_NODES) {
                    float val = acc[t][v] + bn;
                    if (RELU) val = fmaxf(val, 0.0f);
                    out[(size_t)m * DOUT + n] = val;
                }
            }
    }
}

// ---------------------------------------------------------------------------
// global mean pool (sum stage). batch is sorted: blocks of 64 nodes are almost
// always one graph -> LDS tree reduction + 64 atomics; fallback on boundaries.
// ---------------------------------------------------------------------------
__global__ void pool_kernel(const float* __restrict__ h,   // [N, 64]
                            const int* __restrict__ batch,
                            float* __restrict__ pooled) {  // [64, 64]
    const int NB = 64;
    __shared__ float part[256];
    int n0 = blockIdx.x * NB;
    int nEnd = min(n0 + NB, N_NODES);
    int tid = threadIdx.x;
    int g0 = batch[n0];
    int g1 = batch[nEnd - 1];
    if (g0 == g1) {
        int f   = tid & 63;
        int sub = tid >> 6;
        float s = 0.0f;
        for (int nn = n0 + sub; nn < nEnd; nn += 4)
            s += h[(size_t)nn * 64 + f];
        part[tid] = s;
        __syncthreads();
        if (sub == 0) {
            float t = part[f] + part[64 + f] + part[128 + f] + part[192 + f];
            unsafeAtomicAdd(&pooled[(size_t)g0 * 64 + f], t);
        }
    } else {
        for (int i = tid; i < NB * 64; i += 256) {
            int nn = n0 + (i >> 6);
            int f  = i & 63;
            if (nn < nEnd)
                unsafeAtomicAdd(&pooled[(size_t)batch[nn] * 64 + f],
                                h[(size_t)nn * 64 + f]);
        }
    }
}

// counts via binary search on sorted batch, then divide
__global__ void pool_finalize_kernel(const int* __restrict__ batch,
                                     float* __restrict__ pooled) {
    int idx = blockIdx.x * blockDim.x + threadIdx.x;
    if (idx >= N_GRAPHS * 64) return;
    int g = idx >> 6;
    int cnt;
    {
        int lo = 0, hi = N_NODES;
        while (lo < hi) { int mid = (lo + hi) >> 1; if (batch[mid] < g) lo = mid + 1; else hi = mid; }
        int lb = lo;
        lo = 0; hi = N_NODES;
        while (lo < hi) { int mid = (lo + hi) >> 1; if (batch[mid] < g + 1) lo = mid + 1; else hi = mid; }
        cnt = lo - lb;
    }
    pooled[idx] /= fmaxf((float)cnt, 1.0f);
}

// ---------------------------------------------------------------------------
// launcher
// ---------------------------------------------------------------------------
extern "C" void kernel_launch(void* const* d_in, const int* in_sizes, int n_in,
                              void* d_out, int out_size, void* d_ws, size_t ws_size,
                              hipStream_t stream) {
    const float* x     = (const float*)d_in[0];
    const int*   ei    = (const int*)d_in[1];       // [2, E] flat
    const int*   src   = ei;
    const int*   dst   = ei + N_EDGES;
    const int*   batch = (const int*)d_in[2];
    // d_in[3] = edge_attr (unused by reference)
    const float* Wl0 = (const float*)d_in[4];
    const float* Wr0 = (const float*)d_in[5];
    const float* b0  = (const float*)d_in[6];
    const float* Wl1 = (const float*)d_in[7];
    const float* Wr1 = (const float*)d_in[8];
    const float* b1  = (const float*)d_in[9];
    const float* Wl2 = (const float*)d_in[10];
    const float* Wr2 = (const float*)d_in[11];
    const float* b2  = (const float*)d_in[12];
    float* pooled = (float*)d_out;

    // workspace layout (floats)
    float* ws   = (float*)d_ws;
    float* agg  = ws;                               // N*128
    float* bufA = agg  + (size_t)N_NODES * F_IN;    // N*128
    float* bufB = bufA + (size_t)N_NODES * F_IN;    // N*128
    float* deg  = bufB + (size_t)N_NODES * F_IN;    // N
    float* wp0  = deg + N_NODES;                    // 256*128
    float* wp1  = wp0 + 256 * 128;                  // 256*128
    float* wp2  = wp1 + 256 * 128;                  // 256*64

    const int ZB = 2048, ZT = 256;
    const int scatter_blocks = (N_EDGES * 32 + 255) / 256;
    const int gemm_blocks = (N_NODES + 63) / 64;    // 1563

    // one-time prep: degree + packed weights
    zero_kernel<<<256, ZT, 0, stream>>>(deg, (size_t)N_NODES);
    deg_kernel<<<(N_EDGES + 255) / 256, 256, 0, stream>>>(dst, deg, N_EDGES);
    pack_weights_kernel<<<(256 * 128 + 255) / 256, 256, 0, stream>>>(Wl0, Wr0, wp0, 128);
    pack_weights_kernel<<<(256 * 128 + 255) / 256, 256, 0, stream>>>(Wl1, Wr1, wp1, 128);
    pack_weights_kernel<<<(256 * 64  + 255) / 256, 256, 0, stream>>>(Wl2, Wr2, wp2, 64);

    // ---- layer 0: x -> bufA (relu) ----
    zero_kernel<<<ZB, ZT, 0, stream>>>(agg, (size_t)N_NODES * F_IN);
    scatter_kernel<<<scatter_blocks, 256, 0, stream>>>(x, src, dst, agg, N_EDGES);
    sage_gemm_kernel<128, true><<<gemm_blocks, dim3(32, 8), 0, stream>>>(
        agg, x, deg, wp0, b0, bufA);

    // ---- layer 1: bufA -> bufB (relu) ----
    zero_kernel<<<ZB, ZT, 0, stream>>>(agg, (size_t)N_NODES * F_IN);
    scatter_kernel<<<scatter_blocks, 256, 0, stream>>>(bufA, src, dst, agg, N_EDGES);
    sage_gemm_kernel<128, true><<<gemm_blocks, dim3(32, 8), 0, stream>>>(
        agg, bufA, deg, wp1, b1, bufB);

    // ---- layer 2: bufB -> bufA as dense [N,64] (no relu) ----
    zero_kernel<<<ZB, ZT, 0, stream>>>(agg, (size_t)N_NODES * F_IN);
    scatter_kernel<<<scatter_blocks, 256, 0, stream>>>(bufB, src, dst, agg, N_EDGES);
    sage_gemm_kernel<64, false><<<gemm_blocks, dim3(32, 4), 0, stream>>>(
        agg, bufB, deg, wp2, b2, bufA);

    // ---- global mean pool ----
    zero_kernel<<<1, 256, 0, stream>>>(pooled, (size_t)N_GRAPHS * 64);
    pool_kernel<<<(N_NODES + 63) / 64, 256, 0, stream>>>(bufA, batch, pooled);
    pool_finalize_kernel<<<(N_GRAPHS * 64 + 255) / 256, 256, 0, stream>>>(batch, pooled);
}